// QueryExpansionPMField_11519102288258
// MI455X (gfx1250) — compile-verified
//
#include <hip/hip_runtime.h>
#include <math.h>

#define BATCH 4096
#define NCTR  8192
#define DIM   512
#define KEXP  5

#define TM 128
#define TN 128
#define TK 32
#define LSTRIDE 36              // 32 + 4 pad: conflict-free frag reads, 16B-aligned rows
#define STAGES (DIM / TK)       // 16

typedef __attribute__((ext_vector_type(2))) float v2f;
typedef __attribute__((ext_vector_type(8))) float v8f;
typedef __attribute__((ext_vector_type(4))) int   v4i;

#if __has_builtin(__builtin_amdgcn_global_load_async_to_lds_b128)
#define USE_ASYNC 1
typedef __attribute__((address_space(1))) v4i* g_v4i_ptr;   // global (printed as __device__)
typedef __attribute__((address_space(3))) v4i* l_v4i_ptr;   // LDS
__device__ __forceinline__ void wait_async0() {
#if __has_builtin(__builtin_amdgcn_s_wait_asynccnt)
    __builtin_amdgcn_s_wait_asynccnt(0);
#else
    asm volatile("s_wait_asynccnt 0" ::: "memory");
#endif
}
#endif

// ---------------------------------------------------------------------------
// Kernel 0: row sums of squares for z (BATCH rows) and centers (NCTR rows)
// ws[0..BATCH) = ||z_b||^2 ; ws[BATCH..BATCH+NCTR) = ||c_n||^2
// ---------------------------------------------------------------------------
__global__ __launch_bounds__(256)
void pm_norms_kernel(const float* __restrict__ z, const float* __restrict__ cen,
                     float* __restrict__ ws) {
    int lane = threadIdx.x & 31;
    int row  = blockIdx.x * 8 + (threadIdx.x >> 5);
    if (row >= BATCH + NCTR) return;
    const float* p = (row < BATCH) ? (z + (size_t)row * DIM)
                                   : (cen + (size_t)(row - BATCH) * DIM);
    float s = 0.0f;
#pragma unroll
    for (int i = 0; i < 4; ++i) {
        float4 v = *(const float4*)(p + lane * 4 + i * 128);
        s += v.x * v.x + v.y * v.y + v.z * v.z + v.w * v.w;
    }
#pragma unroll
    for (int off = 16; off > 0; off >>= 1) s += __shfl_down(s, off);
    if (lane == 0) ws[row] = s;
}

// ---------------------------------------------------------------------------
// Kernel 1: attractions = mus[n] / (max(zn[m]+cn[n]-2*dot, 0) + 1e-6)
// fp32 WMMA GEMM, 128x128 workgroup tile, 32x64 per wave (8 accumulators)
// double-buffered LDS; async global->LDS staging when available
// ---------------------------------------------------------------------------
__global__ __launch_bounds__(256)
void pm_gemm_kernel(const float* __restrict__ z, const float* __restrict__ cen,
                    const float* __restrict__ mus, const float* __restrict__ ws,
                    float* __restrict__ attr) {
    __shared__ float ldsA[2][TM * LSTRIDE];
    __shared__ float ldsB[2][TN * LSTRIDE];

    const float* znorm = ws;
    const float* cnorm = ws + BATCH;

    const int tid  = threadIdx.x;
    const int wave = tid >> 5;
    const int lane = tid & 31;
    const int lo   = lane & 15;
    const int hi   = lane >> 4;
    const int wm   = wave >> 1;   // 0..3 (M direction, 32 rows each)
    const int wn   = wave & 1;    // 0..1 (N direction, 64 cols each)
    const int m_wg = blockIdx.y * TM;
    const int n_wg = blockIdx.x * TN;

    v8f acc[2][4] = {};

    auto compute_stage = [&](int buf) {
#pragma unroll
        for (int ks = 0; ks < TK; ks += 4) {
            v2f a[2], b[4];
#pragma unroll
            for (int mt = 0; mt < 2; ++mt)
                a[mt] = *(const v2f*)&ldsA[buf][(wm * 32 + mt * 16 + lo) * LSTRIDE + ks + 2 * hi];
#pragma unroll
            for (int nt = 0; nt < 4; ++nt)
                b[nt] = *(const v2f*)&ldsB[buf][(wn * 64 + nt * 16 + lo) * LSTRIDE + ks + 2 * hi];
#pragma unroll
            for (int mt = 0; mt < 2; ++mt)
#pragma unroll
                for (int nt = 0; nt < 4; ++nt)
                    acc[mt][nt] = __builtin_amdgcn_wmma_f32_16x16x4_f32(
                        false, a[mt], false, b[nt], (short)0, acc[mt][nt], false, false);
        }
    };

#if defined(USE_ASYNC)
    // --- async global -> LDS staging, single barrier per stage ---
    auto issue_stage = [&](int s, int buf) {
        const float* za = z   + (size_t)m_wg * DIM + s * TK;
        const float* ca = cen + (size_t)n_wg * DIM + s * TK;
#pragma unroll
        for (int i = 0; i < 4; ++i) {
            int q = tid + i * 256;
            int r = q >> 3, c4 = (q & 7) << 2;
            __builtin_amdgcn_global_load_async_to_lds_b128(
                (g_v4i_ptr)(za + (size_t)r * DIM + c4),
                (l_v4i_ptr)&ldsA[buf][r * LSTRIDE + c4], 0, 0);
            __builtin_amdgcn_global_load_async_to_lds_b128(
                (g_v4i_ptr)(ca + (size_t)r * DIM + c4),
                (l_v4i_ptr)&ldsB[buf][r * LSTRIDE + c4], 0, 0);
        }
    };

    issue_stage(0, 0);
    for (int s = 0; s < STAGES; ++s) {
        wait_async0();                   // own stage-s loads landed in LDS
        __syncthreads();                 // everyone's loads visible; prior readers done
        if (s + 1 < STAGES) issue_stage(s + 1, (s + 1) & 1);
        compute_stage(s & 1);
    }
#else
    // --- fallback: register staging, double-buffered, one barrier per stage ---
    float4 ra[4], rb[4];
    auto load_stage = [&](int s) {
        const float* za = z   + (size_t)m_wg * DIM + s * TK;
        const float* ca = cen + (size_t)n_wg * DIM + s * TK;
#pragma unroll
        for (int i = 0; i < 4; ++i) {
            int q = tid + i * 256;
            int r = q >> 3, c4 = (q & 7) << 2;
            ra[i] = *(const float4*)(za + (size_t)r * DIM + c4);
            rb[i] = *(const float4*)(ca + (size_t)r * DIM + c4);
        }
    };
    auto store_stage = [&](int buf) {
#pragma unroll
        for (int i = 0; i < 4; ++i) {
            int q = tid + i * 256;
            int r = q >> 3, c4 = (q & 7) << 2;
            *(float4*)&ldsA[buf][r * LSTRIDE + c4] = ra[i];
            *(float4*)&ldsB[buf][r * LSTRIDE + c4] = rb[i];
        }
    };
    load_stage(0);
    store_stage(0);
    for (int s = 0; s < STAGES; ++s) {
        if (s + 1 < STAGES) load_stage(s + 1);
        __syncthreads();                 // stage-s LDS stores visible to all waves
        compute_stage(s & 1);
        if (s + 1 < STAGES) store_stage((s + 1) & 1);
    }
#endif

    // epilogue: fold norms + mus, write attractions
#pragma unroll
    for (int nt = 0; nt < 4; ++nt) {
        int n = n_wg + wn * 64 + nt * 16 + lo;
        float cn = cnorm[n];
        float mu = mus[n];
#pragma unroll
        for (int mt = 0; mt < 2; ++mt) {
#pragma unroll
            for (int r = 0; r < 8; ++r) {
                int m = m_wg + wm * 32 + mt * 16 + r + 8 * hi;
                float sq = znorm[m] + cn - 2.0f * acc[mt][nt][r];
                sq = fmaxf(sq, 0.0f) + 1e-6f;
                attr[(size_t)m * NCTR + n] = mu / sq;
            }
        }
    }
}

// ---------------------------------------------------------------------------
// Kernel 2: per-row top-5 -> softmax -> weighted center combine
// one wave per row; register sorted-insert + shfl_xor list merges
// ---------------------------------------------------------------------------
__device__ __forceinline__ void top5_insert(float a, int ai, float v[KEXP], int ix[KEXP]) {
#pragma unroll
    for (int k = 0; k < KEXP; ++k) {
        if (a > v[k]) {
            float tv = v[k]; int ti = ix[k];
            v[k] = a; ix[k] = ai;
            a = tv; ai = ti;
        }
    }
}

__global__ __launch_bounds__(256)
void pm_topk_kernel(const float* __restrict__ z, const float* __restrict__ cen,
                    const float* __restrict__ attr, float* __restrict__ out) {
    int lane = threadIdx.x & 31;
    int row  = blockIdx.x * 8 + (threadIdx.x >> 5);
    const float* arow = attr + (size_t)row * NCTR;

    float v[KEXP]; int ix[KEXP];
#pragma unroll
    for (int k = 0; k < KEXP; ++k) { v[k] = -INFINITY; ix[k] = 0; }

    for (int j = lane; j < NCTR; j += 32) {
        float a = arow[j];
        if (a > v[KEXP - 1]) top5_insert(a, j, v, ix);
    }

#pragma unroll
    for (int off = 16; off > 0; off >>= 1) {
        float ov[KEXP]; int oix[KEXP];
#pragma unroll
        for (int k = 0; k < KEXP; ++k) {
            ov[k]  = __shfl_xor(v[k], off);
            oix[k] = __shfl_xor(ix[k], off);
        }
#pragma unroll
        for (int k = 0; k < KEXP; ++k)
            if (ov[k] > v[KEXP - 1]) top5_insert(ov[k], oix[k], v, ix);
    }
#pragma unroll
    for (int k = 0; k < KEXP; ++k) {
        v[k]  = __shfl(v[k], 0);
        ix[k] = __shfl(ix[k], 0);
    }

    // softmax over the 5 values (v[0] is the max)
    float w[KEXP], s = 0.0f;
#pragma unroll
    for (int k = 0; k < KEXP; ++k) { w[k] = expf(v[k] - v[0]); s += w[k]; }
    float inv = 1.0f / s;
#pragma unroll
    for (int k = 0; k < KEXP; ++k) w[k] *= inv;

    const int d0 = lane * 16;            // 512 dims / 32 lanes
    float comb[16];
#pragma unroll
    for (int q = 0; q < 16; ++q) comb[q] = 0.0f;
#pragma unroll
    for (int k = 0; k < KEXP; ++k) {
        const float* cp = cen + (size_t)ix[k] * DIM + d0;
#pragma unroll
        for (int q = 0; q < 4; ++q) {
            float4 cv = *(const float4*)(cp + q * 4);
            comb[q * 4 + 0] += w[k] * cv.x;
            comb[q * 4 + 1] += w[k] * cv.y;
            comb[q * 4 + 2] += w[k] * cv.z;
            comb[q * 4 + 3] += w[k] * cv.w;
        }
    }
    const float* zr = z + (size_t)row * DIM + d0;
    float* orow = out + (size_t)row * DIM + d0;
#pragma unroll
    for (int q = 0; q < 4; ++q) {
        float4 zv = *(const float4*)(zr + q * 4);
        float4 r;
        r.x = 0.7f * zv.x + 0.3f * comb[q * 4 + 0];
        r.y = 0.7f * zv.y + 0.3f * comb[q * 4 + 1];
        r.z = 0.7f * zv.z + 0.3f * comb[q * 4 + 2];
        r.w = 0.7f * zv.w + 0.3f * comb[q * 4 + 3];
        *(float4*)(orow + q * 4) = r;
    }
}

// ---------------------------------------------------------------------------
extern "C" void kernel_launch(void* const* d_in, const int* in_sizes, int n_in,
                              void* d_out, int out_size, void* d_ws, size_t ws_size,
                              hipStream_t stream) {
    (void)in_sizes; (void)n_in; (void)out_size; (void)ws_size;
    const float* z   = (const float*)d_in[0];
    const float* cen = (const float*)d_in[1];
    const float* mus = (const float*)d_in[2];
    float* out  = (float*)d_out;
    float* expz = out;                                   // (BATCH, DIM)
    float* attr = out + (size_t)BATCH * DIM;             // (BATCH, NCTR)
    float* ws   = (float*)d_ws;                          // norms: BATCH + NCTR floats

    pm_norms_kernel<<<(BATCH + NCTR) / 8, 256, 0, stream>>>(z, cen, ws);

    dim3 g(NCTR / TN, BATCH / TM);                       // 64 x 32 workgroups
    pm_gemm_kernel<<<g, 256, 0, stream>>>(z, cen, mus, ws, attr);

    pm_topk_kernel<<<BATCH / 8, 256, 0, stream>>>(z, cen, attr, expz);
}